// TokenChoiceTopKRouter_2233382993922
// MI455X (gfx1250) — compile-verified
//
#include <hip/hip_runtime.h>
#include <math.h>

typedef __attribute__((ext_vector_type(2))) float v2f;
typedef __attribute__((ext_vector_type(4))) float v4f;
typedef __attribute__((ext_vector_type(8))) float v8f;

constexpr int DIM   = 4096;
constexpr int NE    = 64;
constexpr int TOPK  = 8;
constexpr int NTOK  = 16384;
constexpr int KC    = 128;           // K chunk staged in LDS
constexpr int WAVES = 8;             // waves per block
constexpr int BLKT  = WAVES * 16;    // 128 tokens per block
constexpr int TK    = NTOK * TOPK;   // 131072 routed slots
constexpr int SORTB = 64;            // sort blocks
constexpr int CHUNK = TK / SORTB;    // 2048 slots per sort block

// ---------------------------------------------------------------------------
// Kernel 1: router GEMM (fp32 WMMA 16x16x4) + bias + top-8 + softmax
// Each wave: 16 tokens x 64 experts. acc[nt] lane layout (ISA 7.12.2):
//   expert e = nt*16 + (lane&15), token m = vgpr_r + 8*(lane>>4)
// A frag (16x4 f32): lane = (m = lane&15), VGPR0/1 = K = 2*(lane>>4)+{0,1}
// B frag (4x16 f32): lane = (n = lane&15), VGPR0/1 = K = 2*(lane>>4)+{0,1}
// ---------------------------------------------------------------------------
__global__ __launch_bounds__(256, 2)
void router_gemm_topk(const float* __restrict__ x, const float* __restrict__ gw,
                      const float* __restrict__ gb, const float* __restrict__ eb,
                      float* __restrict__ ws_scores, int* __restrict__ ws_experts)
{
  __shared__ float ldsW[KC / 2][NE][2];      // [k-pair][expert][2]  = 32 KB
  __shared__ float ldsLog[WAVES][16][NE];    // per-wave logits      = 32 KB

  const int tid   = threadIdx.x;
  const int wave  = tid >> 5;
  const int lane  = tid & 31;
  const int m     = lane & 15;
  const int khalf = lane >> 4;
  const int tokBase = blockIdx.x * BLKT + wave * 16;

  v8f acc[4] = {};

  for (int kc = 0; kc < DIM; kc += KC) {
    __syncthreads();
    // cooperative stage of gate_w chunk: 64 experts x 128 k, b128 loads
    for (int i = tid; i < NE * (KC / 4); i += 256) {
      const int e = i >> 5;      // KC/4 = 32 float4 per expert row
      const int j = i & 31;
      v4f w4 = *(const v4f*)(gw + (size_t)e * DIM + kc + 4 * j);
      ldsW[2 * j    ][e][0] = w4.x; ldsW[2 * j    ][e][1] = w4.y;
      ldsW[2 * j + 1][e][0] = w4.z; ldsW[2 * j + 1][e][1] = w4.w;
    }
    __syncthreads();

    const float* xrow = x + (size_t)(tokBase + m) * DIM + kc + 2 * khalf;
    if (kc + KC < DIM) __builtin_prefetch(xrow + KC, 0, 0);  // global_prefetch_b8

    #pragma unroll 8
    for (int ks = 0; ks < KC / 4; ++ks) {
      v2f a = *(const v2f*)(xrow + 4 * ks);          // K = 4*ks + 2*khalf + {0,1}
      const int p = 2 * ks + khalf;
      #pragma unroll
      for (int nt = 0; nt < 4; ++nt) {
        v2f b = *(const v2f*)&ldsW[p][nt * 16 + m][0];
        acc[nt] = __builtin_amdgcn_wmma_f32_16x16x4_f32(
            false, a, false, b, (short)0, acc[nt], false, false);
      }
    }
  }

  // bias (gate_b + expert_bias) depends only on expert column
  #pragma unroll
  for (int nt = 0; nt < 4; ++nt) {
    const int e = nt * 16 + m;
    const float bias = gb[e] + eb[e];
    #pragma unroll
    for (int r = 0; r < 8; ++r) acc[nt][r] += bias;
  }

  // transpose logits through LDS: ldsLog[wave][token][expert]
  #pragma unroll
  for (int nt = 0; nt < 4; ++nt)
    #pragma unroll
    for (int r = 0; r < 8; ++r)
      ldsLog[wave][r + 8 * khalf][nt * 16 + m] = acc[nt][r];
  __syncthreads();

  // lanes 0..15: one token each -> register-resident top-8 + softmax
  if (lane < 16) {
    const float* lg = &ldsLog[wave][lane][0];
    float s[TOPK];
    int   id[TOPK];
    #pragma unroll
    for (int k = 0; k < TOPK; ++k) { s[k] = -INFINITY; id[k] = 0; }

    for (int e = 0; e < NE; ++e) {
      float v = lg[e];
      int   vi = e;
      #pragma unroll
      for (int k = 0; k < TOPK; ++k) {     // descending; ties keep lowest index
        const bool gt = v > s[k];
        const float ns = gt ? v : s[k];
        const int  nid = gt ? vi : id[k];
        v  = gt ? s[k]  : v;
        vi = gt ? id[k] : vi;
        s[k] = ns; id[k] = nid;
      }
    }

    const float mx = s[0];
    float p[TOPK], sum = 0.f;
    #pragma unroll
    for (int k = 0; k < TOPK; ++k) { p[k] = __expf(s[k] - mx); sum += p[k]; }
    const float inv = 1.0f / sum;

    const int tt = tokBase + lane;
    #pragma unroll
    for (int k = 0; k < TOPK; ++k) {
      ws_scores[tt * TOPK + k]  = p[k] * inv;
      ws_experts[tt * TOPK + k] = id[k];
    }
  }
}

// ---------------------------------------------------------------------------
// Kernel 2: per-block expert histogram (counting-sort pass 1)
// ---------------------------------------------------------------------------
__global__ __launch_bounds__(256)
void block_hist(const int* __restrict__ ex, int* __restrict__ bc)
{
  __shared__ int h[NE];
  const int tid = threadIdx.x;
  if (tid < NE) h[tid] = 0;
  __syncthreads();
  const int base = blockIdx.x * CHUNK;
  for (int i = tid; i < CHUNK; i += 256) atomicAdd(&h[ex[base + i]], 1);
  __syncthreads();
  if (tid < NE) bc[blockIdx.x * NE + tid] = h[tid];
}

// ---------------------------------------------------------------------------
// Kernel 3: bin totals -> num_tokens_per_expert; exclusive scan over bins;
// column-wise scan over blocks -> per-block start offsets (in-place in bc)
// ---------------------------------------------------------------------------
__global__ __launch_bounds__(64)
void scan_offsets(int* __restrict__ bc, int* __restrict__ hist)
{
  __shared__ int tot[NE];
  const int e = threadIdx.x;
  int s = 0;
  for (int b = 0; b < SORTB; ++b) s += bc[b * NE + e];
  tot[e] = s;
  hist[e] = s;                         // num_tokens_per_expert
  __syncthreads();
  int run = 0;
  for (int i = 0; i < NE; ++i) { if (i == e) break; run += tot[i]; }  // exclusive scan
  for (int b = 0; b < SORTB; ++b) {
    const int c = bc[b * NE + e];
    bc[b * NE + e] = run;
    run += c;
  }
}

// ---------------------------------------------------------------------------
// Kernel 4: stable scatter. One wave per block; in-order 32-slot groups.
// Rank within a group via 6-bit ballot match; LDS running counters give
// cross-group stability (wave32 program order -> no atomics needed).
// ---------------------------------------------------------------------------
__global__ __launch_bounds__(32)
void scatter_sorted(const float* __restrict__ sc, const int* __restrict__ ex,
                    const int* __restrict__ bc,
                    float* __restrict__ out_sc, int* __restrict__ out_ti)
{
  __shared__ int ctr[NE];
  const int lane = threadIdx.x;
  ctr[lane]      = bc[blockIdx.x * NE + lane];
  ctr[lane + 32] = bc[blockIdx.x * NE + lane + 32];
  __syncthreads();

  const int base = blockIdx.x * CHUNK;
  const unsigned below = (1u << lane) - 1u;

  for (int g = 0; g < CHUNK / 32; ++g) {
    const int slot = base + g * 32 + lane;
    const int e = ex[slot];
    unsigned same = 0xFFFFFFFFu;
    #pragma unroll
    for (int bit = 0; bit < 6; ++bit) {
      const unsigned bm = (unsigned)__ballot((e >> bit) & 1);
      same &= ((e >> bit) & 1) ? bm : ~bm;
    }
    const int rank = __popc(same & below);
    const int cnt  = __popc(same);
    const int pos  = ctr[e] + rank;          // all lanes read before any write
    out_sc[pos] = sc[slot];
    out_ti[pos] = slot >> 3;                 // order // TOP_K
    if (rank == cnt - 1) ctr[e] = pos + 1;   // one writer per expert
  }
}

// ---------------------------------------------------------------------------
extern "C" void kernel_launch(void* const* d_in, const int* in_sizes, int n_in,
                              void* d_out, int out_size, void* d_ws, size_t ws_size,
                              hipStream_t stream)
{
  const float* x  = (const float*)d_in[0];
  const float* gw = (const float*)d_in[1];
  const float* gb = (const float*)d_in[2];
  const float* eb = (const float*)d_in[3];

  float* out_scores = (float*)d_out;               // [TK] f32
  int*   out_tidx   = (int*)d_out + TK;            // [TK] i32 (bit-cast region)
  int*   out_hist   = (int*)d_out + 2 * TK;        // [NE] i32

  float* ws_scores  = (float*)d_ws;                // [TK] f32
  int*   ws_experts = (int*)d_ws + TK;             // [TK] i32
  int*   bc         = (int*)d_ws + 2 * TK;         // [SORTB][NE] i32

  router_gemm_topk<<<NTOK / BLKT, 256, 0, stream>>>(x, gw, gb, eb, ws_scores, ws_experts);
  block_hist<<<SORTB, 256, 0, stream>>>(ws_experts, bc);
  scan_offsets<<<1, 64, 0, stream>>>(bc, out_hist);
  scatter_sorted<<<SORTB, 32, 0, stream>>>(ws_scores, ws_experts, bc, out_scores, out_tidx);
}